// BackgroundStd2D_11476152615311
// MI455X (gfx1250) — compile-verified
//
#include <hip/hip_runtime.h>
#include <hip/hip_bf16.h>
#include <stdint.h>

// ---------------------------------------------------------------------------
// BackgroundStd2D: masked per-(b,c) std over H*W, single pass.
//   bf:        (4, 256, 256, 256) f32   -> 256 MiB streamed once (HBM bound)
//   aspp_mask: (4, 1, 256, 256)   f32   -> packed to bits in stage 1
//   min_std:   (1, 256, 1, 1)     f32
//   out:       (4, 256, 1, 1)     f32   (1024 floats)
// ---------------------------------------------------------------------------

#define HW        65536      // 256*256 pixels per plane
#define THREADS   256        // 8 wave32 waves
#define CHUNK     8192       // floats per LDS chunk (32 KB)
#define NCHUNK    (HW / CHUNK)           // 8
#define QPC       (CHUNK / (THREADS*4))  // float4 quads per thread per chunk = 8
#define BITS_PER_BATCH 2048  // dwords: 65536 bits / 32

typedef __attribute__((address_space(3))) float       lds_f32;
typedef __attribute__((address_space(1))) const float gbl_cf32;

#if __has_builtin(__builtin_amdgcn_s_wait_asynccnt)
#define WAIT_ASYNC(n) __builtin_amdgcn_s_wait_asynccnt(n)
#else
#define WAIT_ASYNC(n) asm volatile("s_wait_asynccnt %0" :: "i"(n) : "memory")
#endif

// One lane moves 16 B global -> LDS, tracked by ASYNCcnt (no VGPR dest, so we
// can keep far more bytes in flight than LOADcnt/VGPR-destination loads allow).
__device__ __forceinline__ void async_ld_b128(float* lds_dst, const float* gsrc) {
    asm volatile("global_load_async_to_lds_b128 %0, %1, off"
                 :: "v"((lds_f32*)lds_dst), "v"((gbl_cf32*)gsrc)
                 : "memory");
}

// Issue a whole 32 KB chunk: 8 async b128 per thread (block moves 4 KB/instr).
__device__ __forceinline__ void issue_chunk(const float* gbase, float* lbase, int t) {
#pragma unroll
    for (int q = 0; q < QPC; ++q) {
        const int e = (q * THREADS + t) * 4;          // float index in chunk
        async_ld_b128(lbase + e, gbase + e);
    }
}

// ---------------------------------------------------------------------------
// Stage 1: pack keep = (mask <= 0.5) into bits, in stage-2 per-thread order.
// Thread t of batch-block b owns quads (i*256+t), i=0..63; nibble for quad i
// lives in dword (i>>3), nibble slot (i&7). Stored at bits[b*2048 + t*8 + k].
// ---------------------------------------------------------------------------
__global__ __launch_bounds__(THREADS) void mask_pack_kernel(
        const float* __restrict__ mask, uint32_t* __restrict__ bits) {
    const int b = blockIdx.x;
    const int t = threadIdx.x;
    const float4* mp = (const float4*)(mask + (size_t)b * HW);

    uint32_t w[8];
#pragma unroll
    for (int k = 0; k < 8; ++k) w[k] = 0u;

#pragma unroll
    for (int i = 0; i < 64; ++i) {
        float4 m = mp[i * THREADS + t];
        uint32_t nib = (m.x <= 0.5f ? 1u : 0u)
                     | (m.y <= 0.5f ? 2u : 0u)
                     | (m.z <= 0.5f ? 4u : 0u)
                     | (m.w <= 0.5f ? 8u : 0u);
        w[i >> 3] |= nib << ((i & 7) * 4);
    }

    uint32_t* dst = bits + (size_t)b * BITS_PER_BATCH + t * 8;
    *(uint4*)(dst + 0) = make_uint4(w[0], w[1], w[2], w[3]);
    *(uint4*)(dst + 4) = make_uint4(w[4], w[5], w[6], w[7]);
}

// ---------------------------------------------------------------------------
// Stage 2: one block per (b,c) plane. Double-buffered async global->LDS
// streaming of bf; accumulate n (via popcount), s = sum(x*k), ss = sum(x^2*k).
// std = sqrt((ss - s^2/n) / (n-1));  out = max(std, 1e-6 + min_std[c]).
// ---------------------------------------------------------------------------
__global__ __launch_bounds__(THREADS) void bg_std_kernel(
        const float* __restrict__ bf, const uint32_t* __restrict__ bits,
        const float* __restrict__ min_std, float* __restrict__ out) {
    __shared__ __align__(16) float lbuf[2][CHUNK];   // 64 KB double buffer

    const int t   = threadIdx.x;
    const int blk = blockIdx.x;                      // blk = b*256 + c
    const int b   = blk >> 8;
    const int c   = blk & 255;
    const float* plane = bf + (size_t)blk * HW;

    // This thread's 256 mask bits (8 dwords), packed by stage 1.
    const uint32_t* bp = bits + (size_t)b * BITS_PER_BATCH + t * 8;
    uint4 blo = *(const uint4*)(bp + 0);
    uint4 bhi = *(const uint4*)(bp + 4);
    uint32_t bw[8] = {blo.x, blo.y, blo.z, blo.w, bhi.x, bhi.y, bhi.z, bhi.w};

    float cnt = (float)(__popc(bw[0]) + __popc(bw[1]) + __popc(bw[2]) + __popc(bw[3]) +
                        __popc(bw[4]) + __popc(bw[5]) + __popc(bw[6]) + __popc(bw[7]));

    float s = 0.0f, ss = 0.0f;

    issue_chunk(plane, &lbuf[0][0], t);              // prime buffer 0
#pragma unroll
    for (int ck = 0; ck < NCHUNK; ++ck) {
        if (ck + 1 < NCHUNK) {
            issue_chunk(plane + (ck + 1) * CHUNK, &lbuf[(ck + 1) & 1][0], t);
            WAIT_ASYNC(QPC);   // loads retire in order: <=8 pending => chunk ck done
        } else {
            WAIT_ASYNC(0);
        }
        const float* lb = &lbuf[ck & 1][0];
        const uint32_t w = bw[ck];                   // dword ck <-> iterations ck*8..+7
#pragma unroll
        for (int j = 0; j < QPC; ++j) {
            float4 v = *(const float4*)(lb + (j * THREADS + t) * 4);
            uint32_t nib = w >> (j * 4);
            float x0 = (nib & 1u) ? v.x : 0.0f;
            float x1 = (nib & 2u) ? v.y : 0.0f;
            float x2 = (nib & 4u) ? v.z : 0.0f;
            float x3 = (nib & 8u) ? v.w : 0.0f;
            s += (x0 + x1) + (x2 + x3);
            ss = fmaf(x0, x0, fmaf(x1, x1, fmaf(x2, x2, fmaf(x3, x3, ss))));
        }
    }

    // wave32 butterfly reduction
#pragma unroll
    for (int off = 16; off > 0; off >>= 1) {
        s   += __shfl_xor(s,   off, 32);
        ss  += __shfl_xor(ss,  off, 32);
        cnt += __shfl_xor(cnt, off, 32);
    }

    // cross-wave reduction (reuse lbuf; all async traffic already retired)
    const int wave = t >> 5, lane = t & 31;
    __syncthreads();
    if (lane == 0) {
        lbuf[0][wave]      = s;
        lbuf[0][8 + wave]  = ss;
        lbuf[0][16 + wave] = cnt;
    }
    __syncthreads();
    if (t == 0) {
        float S = 0.0f, SS = 0.0f, N = 0.0f;
#pragma unroll
        for (int wv = 0; wv < THREADS / 32; ++wv) {
            S  += lbuf[0][wv];
            SS += lbuf[0][8 + wv];
            N  += lbuf[0][16 + wv];
        }
        float var = (SS - S * S / N) / (N - 1.0f);
        float sd  = sqrtf(fmaxf(var, 0.0f));
        out[blk]  = fmaxf(sd, 1e-6f + min_std[c]);
    }
}

// ---------------------------------------------------------------------------
extern "C" void kernel_launch(void* const* d_in, const int* in_sizes, int n_in,
                              void* d_out, int out_size, void* d_ws, size_t ws_size,
                              hipStream_t stream) {
    const float* bf      = (const float*)d_in[0];   // (4,256,256,256) f32
    const float* mask    = (const float*)d_in[1];   // (4,1,256,256)   f32
    const float* min_std = (const float*)d_in[2];   // (1,256,1,1)     f32
    float*       out     = (float*)d_out;           // 1024 f32
    uint32_t*    bits    = (uint32_t*)d_ws;         // 4 * 2048 dwords = 32 KB

    mask_pack_kernel<<<dim3(4),    dim3(THREADS), 0, stream>>>(mask, bits);
    bg_std_kernel  <<<dim3(1024),  dim3(THREADS), 0, stream>>>(bf, bits, min_std, out);
}